// CustomBigBirdAttention_18356690223152
// MI455X (gfx1250) — compile-verified
//
#include <hip/hip_runtime.h>
#include <hip/hip_bf16.h>

// CDNA5 / gfx1250 wave32 WMMA implementation of BigBird sparse attention fwd.
// All GEMMs run on V_WMMA_F32_16X16X4_F32 (full fp32, matches reference math).
// Softmax is fused into the context kernel via LDS-staged exp(probs) chunks.

typedef __attribute__((ext_vector_type(2))) float v2f;
typedef __attribute__((ext_vector_type(8))) float v8f;

constexpr int SEQ   = 768;   // S
constexpr int DMODEL= 256;   // D
constexpr int NH    = 8;     // H
constexpr int KDIM  = 256;   // KD
constexpr int HDIM  = 32;    // DH
constexpr int BLK   = 64;    // BS
constexpr int CHUNK = 384;   // M-chunk staged in LDS for the context kernel
constexpr int STR   = CHUNK + 4;  // LDS row stride (bank-conflict padding)
constexpr float QSCALE = 0.17677669529663688f;  // 32^-0.5

__device__ __forceinline__ v8f wmma_f32_k4(v2f a, v2f b, v8f c) {
  // 8 args: (neg_a, A, neg_b, B, c_mod, C, reuse_a, reuse_b)
  return __builtin_amdgcn_wmma_f32_16x16x4_f32(false, a, false, b, (short)0, c,
                                               false, false);
}

// ---------------------------------------------------------------------------
// Kernel A: fused QKV projection. One wave per 16x32 output strip
// (two accumulators sharing one A fragment per K-step).
//   z = 0: q = (query @ Wq + bq) * QSCALE   -> q_ws[H][S][32]
//   z = 1: k =  key_in @ Wk + bk            -> k_ws[H][S][32]
//   z = 2: v =  value  @ Wv + bv            -> v_ws[H][S][32]
// ---------------------------------------------------------------------------
__global__ void qkv_proj_kernel(const float* __restrict__ query,
                                const float* __restrict__ key_in,
                                const float* __restrict__ value,
                                const float* __restrict__ Wq, const float* __restrict__ bq,
                                const float* __restrict__ Wk, const float* __restrict__ bk,
                                const float* __restrict__ Wv, const float* __restrict__ bv,
                                float* __restrict__ q_ws,
                                float* __restrict__ k_ws,
                                float* __restrict__ v_ws) {
  const int lane  = threadIdx.x & 31;
  const int col0  = blockIdx.x * 32;     // 32-wide output strip
  const int row0  = blockIdx.y * 16;
  const int which = blockIdx.z;

  const float* X    = (which == 0) ? query : (which == 1) ? key_in : value;
  const float* W    = (which == 0) ? Wq    : (which == 1) ? Wk     : Wv;
  const float* bias = (which == 0) ? bq    : (which == 1) ? bk     : bv;
  float*       out  = (which == 0) ? q_ws  : (which == 1) ? k_ws   : v_ws;

  const int half = lane >> 4;      // 0 -> K{0,1}, 1 -> K{2,3}
  const int l16  = lane & 15;
  const int arow  = row0 + l16;
  const int bcol0 = col0 + l16;
  const int bcol1 = col0 + 16 + l16;

  v8f acc0 = {}, acc1 = {};
#pragma unroll 4
  for (int k = 0; k < DMODEL; k += 4) {
    const int ka = k + half * 2;
    v2f a, b0, b1;
    a.x  = X[arow * DMODEL + ka];
    a.y  = X[arow * DMODEL + ka + 1];
    b0.x = W[(ka)     * KDIM + bcol0];
    b0.y = W[(ka + 1) * KDIM + bcol0];
    b1.x = W[(ka)     * KDIM + bcol1];
    b1.y = W[(ka + 1) * KDIM + bcol1];
    acc0 = wmma_f32_k4(a, b0, acc0);
    acc1 = wmma_f32_k4(a, b1, acc1);
  }

  const float scale = (which == 0) ? QSCALE : 1.0f;
  const float bc0 = bias[bcol0], bc1 = bias[bcol1];
  const int h0 = bcol0 >> 5, dh0 = bcol0 & 31;
  const int h1 = bcol1 >> 5, dh1 = bcol1 & 31;
#pragma unroll
  for (int r = 0; r < 8; ++r) {
    const int row = row0 + r + half * 8;
    out[(h0 * SEQ + row) * HDIM + dh0] = (acc0[r] + bc0) * scale;
    out[(h1 * SEQ + row) * HDIM + dh1] = (acc1[r] + bc1) * scale;
  }
}

// ---------------------------------------------------------------------------
// Kernel B: scores[h][s][m] = q[h][s] . k[h][idx(m)] + mask[s][m]
// One workgroup (8 waves) per (slot, qblock, head) -> 64x64 score tile.
// Each wave owns one 16-row strip x two 16-col tiles; the Q (A) fragment is
// loaded once per K-step and shared across both accumulators.
// ---------------------------------------------------------------------------
__global__ void scores_kernel(const float* __restrict__ q_ws,
                              const float* __restrict__ k_ws,
                              const int*   __restrict__ attn_indices,
                              const float* __restrict__ blocked_mask,
                              float* __restrict__ scores, int M) {
  const int lane = threadIdx.x & 31;
  const int wave = threadIdx.x >> 5;        // 0..7
  const int slot = blockIdx.x;              // 64-wide column slot within M
  const int qb   = blockIdx.y;              // query block (64 rows)
  const int h    = blockIdx.z;

  const int half = lane >> 4;
  const int l16  = lane & 15;

  const int tr  = wave >> 1;                // row strip 0..3
  const int tc0 = (wave & 1) * 2;           // first of two col tiles
  const int row0 = qb * BLK + tr * 16;
  const int colA = slot * BLK + tc0 * 16 + l16;
  const int colB = colA + 16;

  const long idx_base = (long)(qb * BLK) * M;   // indices shared across block
  const int kidx0 = attn_indices[idx_base + colA];
  const int kidx1 = attn_indices[idx_base + colB];

  const float* qrow  = q_ws + (long)(h * SEQ + row0 + l16) * HDIM;
  const float* krow0 = k_ws + (long)(h * SEQ + kidx0)      * HDIM;
  const float* krow1 = k_ws + (long)(h * SEQ + kidx1)      * HDIM;

  v8f acc0 = {}, acc1 = {};
#pragma unroll
  for (int kk = 0; kk < HDIM; kk += 4) {
    const int ko = kk + half * 2;
    v2f a, b0, b1;
    a.x  = qrow[ko];   a.y  = qrow[ko + 1];
    b0.x = krow0[ko];  b0.y = krow0[ko + 1];
    b1.x = krow1[ko];  b1.y = krow1[ko + 1];
    acc0 = wmma_f32_k4(a, b0, acc0);
    acc1 = wmma_f32_k4(a, b1, acc1);
  }

#pragma unroll
  for (int r = 0; r < 8; ++r) {
    const int row = row0 + r + half * 8;
    scores[((long)(h * SEQ + row)) * M + colA] =
        acc0[r] + blocked_mask[(long)row * M + colA];
    scores[((long)(h * SEQ + row)) * M + colB] =
        acc1[r] + blocked_mask[(long)row * M + colB];
  }
}

// ---------------------------------------------------------------------------
// Kernel C: fused softmax + context.
//   ctx[s][h*32+dh] = (1/rowsum) * sum_m exp(scores[h][s][m]-rowmax) * v[h][idx(m)][dh]
// One workgroup (8 waves, 256 threads) per (qblock, head).
//   Phase 1: wave-per-row shuffle reductions -> rowmax, 1/rowsum (LDS).
//   Phase 2: stage exp(s - rowmax) for a 64xCHUNK block into LDS (padded
//            stride kills A-read bank conflicts), then WMMA with A from LDS
//            and gathered V rows from global; accumulate across chunks.
//   The 1/rowsum scale is applied once at writeout (softmax is linear in ctx).
// ---------------------------------------------------------------------------
__global__ void context_kernel(const float* __restrict__ scores,
                               const float* __restrict__ v_ws,
                               const int*   __restrict__ attn_indices,
                               float* __restrict__ ctx, int M) {
  extern __shared__ char smem[];
  float* sp     = (float*)smem;                                  // [64][STR]
  int*   lidx   = (int*)(smem + (size_t)BLK * STR * 4);          // [768]
  float* rmax_s = (float*)(smem + (size_t)BLK * STR * 4 + 768 * 4);  // [64]
  float* rinv_s = rmax_s + BLK;                                  // [64]

  const int lane = threadIdx.x & 31;
  const int wave = threadIdx.x >> 5;        // 0..7
  const int qb   = blockIdx.x;
  const int h    = blockIdx.y;

  // --- stage gather indices (shared by all 64 rows of this block) ---
  const long idx_base = (long)(qb * BLK) * M;
  for (int m = threadIdx.x; m < M && m < 768; m += blockDim.x)
    lidx[m] = attn_indices[idx_base + m];

  // --- phase 1: per-row max and sum (wave-per-row, shuffle reductions) ---
  for (int i = 0; i < 8; ++i) {
    const int r = wave * 8 + i;             // local row 0..63
    const float* srow = scores + ((long)(h * SEQ + qb * BLK + r)) * M;
    float lm = -3.4e38f;
    for (int m = lane; m < M; m += 32) lm = fmaxf(lm, srow[m]);
#pragma unroll
    for (int off = 16; off > 0; off >>= 1) lm = fmaxf(lm, __shfl_xor(lm, off, 32));
    float ls = 0.f;
    for (int m = lane; m < M; m += 32) ls += __expf(srow[m] - lm);
#pragma unroll
    for (int off = 16; off > 0; off >>= 1) ls += __shfl_xor(ls, off, 32);
    if (lane == 0) { rmax_s[r] = lm; rinv_s[r] = 1.0f / ls; }
  }
  __syncthreads();

  // --- phase 2: chunked exp-staging + WMMA ---
  const int half = lane >> 4;
  const int l16  = lane & 15;
  const int tr   = wave >> 1, tc = wave & 1;      // 4x2 tiles of 16
  const int row0l = tr * 16;                      // local row of this tile
  const int dcol  = tc * 16 + l16;                // 0..31 within head dim
  const float* vh = v_ws + (long)h * SEQ * HDIM;

  v8f acc = {};
  for (int c0 = 0; c0 < M; c0 += CHUNK) {
    // stage exp(s - rowmax) for cols [c0, c0+CHUNK)
    for (int i = 0; i < 8; ++i) {
      const int r = wave * 8 + i;
      const float* srow = scores + ((long)(h * SEQ + qb * BLK + r)) * M + c0;
      const float rm = rmax_s[r];
      for (int m = lane; m < CHUNK; m += 32)
        sp[r * STR + m] = __expf(srow[m] - rm);
    }
    __syncthreads();

    for (int mm = 0; mm < CHUNK; mm += 4) {
      const int mo = mm + half * 2;
      v2f a, b;
      a.x = sp[(row0l + l16) * STR + mo];
      a.y = sp[(row0l + l16) * STR + mo + 1];
      b.x = vh[(long)lidx[c0 + mo]     * HDIM + dcol];
      b.y = vh[(long)lidx[c0 + mo + 1] * HDIM + dcol];
      acc = wmma_f32_k4(a, b, acc);
    }
    __syncthreads();
  }

#pragma unroll
  for (int r = 0; r < 8; ++r) {
    const int rl  = row0l + r + half * 8;         // local row 0..63
    const int row = qb * BLK + rl;
    ctx[(long)row * KDIM + h * HDIM + dcol] = acc[r] * rinv_s[rl];
  }
}

// ---------------------------------------------------------------------------
// Kernel D: out = ctx @ Wo + bo. One wave per 16x32 output strip.
// ---------------------------------------------------------------------------
__global__ void out_proj_kernel(const float* __restrict__ ctx,
                                const float* __restrict__ Wo,
                                const float* __restrict__ bo,
                                float* __restrict__ out) {
  const int lane = threadIdx.x & 31;
  const int col0 = blockIdx.x * 32;
  const int row0 = blockIdx.y * 16;
  const int half = lane >> 4;
  const int l16  = lane & 15;
  const int arow  = row0 + l16;
  const int bcol0 = col0 + l16;
  const int bcol1 = col0 + 16 + l16;

  v8f acc0 = {}, acc1 = {};
#pragma unroll 4
  for (int k = 0; k < KDIM; k += 4) {
    const int ka = k + half * 2;
    v2f a, b0, b1;
    a.x  = ctx[arow * KDIM + ka];
    a.y  = ctx[arow * KDIM + ka + 1];
    b0.x = Wo[(ka)     * DMODEL + bcol0];
    b0.y = Wo[(ka + 1) * DMODEL + bcol0];
    b1.x = Wo[(ka)     * DMODEL + bcol1];
    b1.y = Wo[(ka + 1) * DMODEL + bcol1];
    acc0 = wmma_f32_k4(a, b0, acc0);
    acc1 = wmma_f32_k4(a, b1, acc1);
  }

  const float bc0 = bo[bcol0], bc1 = bo[bcol1];
#pragma unroll
  for (int r = 0; r < 8; ++r) {
    const int row = row0 + r + half * 8;
    out[row * DMODEL + bcol0] = acc0[r] + bc0;
    out[row * DMODEL + bcol1] = acc1[r] + bc1;
  }
}

// Kernel E: third tuple output — attn_indices as float values.
__global__ void idx_copy_kernel(const int* __restrict__ idx,
                                float* __restrict__ dst, int n) {
  const int i = blockIdx.x * blockDim.x + threadIdx.x;
  if (i < n) dst[i] = (float)idx[i];
}

// ---------------------------------------------------------------------------
extern "C" void kernel_launch(void* const* d_in, const int* in_sizes, int n_in,
                              void* d_out, int out_size, void* d_ws, size_t ws_size,
                              hipStream_t stream) {
  const float* query        = (const float*)d_in[0];
  const float* value        = (const float*)d_in[1];
  const float* key_in       = (const float*)d_in[2];
  const float* Wq           = (const float*)d_in[3];
  const float* bq           = (const float*)d_in[4];
  const float* Wk           = (const float*)d_in[5];
  const float* bk           = (const float*)d_in[6];
  const float* Wv           = (const float*)d_in[7];
  const float* bv           = (const float*)d_in[8];
  const float* Wo           = (const float*)d_in[9];
  const float* bo           = (const float*)d_in[10];
  const int*   attn_indices = (const int*)d_in[11];
  const float* blocked_mask = (const float*)d_in[12];

  const int M = in_sizes[11] / SEQ;   // padded neighbor width (768 here)

  // Workspace layout (floats): q/k/v head-split + ctx
  float* q_ws = (float*)d_ws;
  float* k_ws = q_ws + NH * SEQ * HDIM;
  float* v_ws = k_ws + NH * SEQ * HDIM;
  float* ctx  = v_ws + NH * SEQ * HDIM;

  // Output layout: [out | scores | attn_indices(float)]
  float* out0    = (float*)d_out;
  float* scoresO = out0 + SEQ * DMODEL;
  float* idxO    = scoresO + (size_t)NH * SEQ * M;

  // A: QKV projections (wave per 16x32 strip, shared-A WMMA)
  qkv_proj_kernel<<<dim3(KDIM / 32, SEQ / 16, 3), 32, 0, stream>>>(
      query, key_in, value, Wq, bq, Wk, bk, Wv, bv, q_ws, k_ws, v_ws);

  // B: gathered-K scores + mask
  scores_kernel<<<dim3(M / BLK, SEQ / BLK, NH), 256, 0, stream>>>(
      q_ws, k_ws, attn_indices, blocked_mask, scoresO, M);

  // C: fused softmax + gathered-V context (LDS-staged probs chunks)
  const size_t shmem = (size_t)BLK * STR * 4 + 768 * 4 + 2 * BLK * 4;
  context_kernel<<<dim3(SEQ / BLK, NH), 256, shmem, stream>>>(
      scoresO, v_ws, attn_indices, ctx, M);

  // D: output projection
  out_proj_kernel<<<dim3(DMODEL / 32, SEQ / 16), 32, 0, stream>>>(
      ctx, Wo, bo, out0);

  // E: attn_indices tuple output
  const int n_idx = SEQ * M;
  idx_copy_kernel<<<(n_idx + 255) / 256, 256, 0, stream>>>(attn_indices, idxO, n_idx);
}